// OnlineCalibratorInfer_80290118632068
// MI455X (gfx1250) — compile-verified
//
#include <hip/hip_runtime.h>

typedef float v2f __attribute__((ext_vector_type(2)));
typedef float v8f __attribute__((ext_vector_type(8)));

#define M_TOT 16384   // B*S = 4*4096
#define N_TOT 4096
#define K_TOT 4096
#define BM 128
#define BN 128
#define BK 16
#define LDA 20    // padded LDS stride for A tile (dwords): 80B rows, 16B-aligned, conflict-free frag reads
#define LDB 136   // padded LDS stride for B tile (dwords): 544B rows, halves of wave hit disjoint banks

#if defined(__AMDGCN__) && \
    __has_builtin(__builtin_amdgcn_global_load_async_to_lds_b128) && \
    __has_builtin(__builtin_amdgcn_s_wait_asynccnt)
#define USE_ASYNC_LDS 1
// Builtin signature (from hipcc diagnostic): param0 is
//   '__attribute__((__vector_size__(4*sizeof(int)))) int __device__ *'  (int4* in AS1/global)
// so pass exactly-typed int4 pointers in AS1 (src) and AS3 (LDS dst).
typedef int v4i_vs __attribute__((vector_size(4 * sizeof(int))));
typedef __attribute__((address_space(1))) void    as1_void;
typedef __attribute__((address_space(3))) void    as3_void;
typedef __attribute__((address_space(1))) v4i_vs  as1_v4i;
typedef __attribute__((address_space(3))) v4i_vs  as3_v4i;
#else
#define USE_ASYNC_LDS 0
#endif

__global__ __launch_bounds__(256)
void rotate_smooth_gemm_f32(const float* __restrict__ x,
                            const float* __restrict__ rot,
                            const float* __restrict__ smooth,
                            float* __restrict__ out) {
    __shared__ float As[2][BM * LDA];
    __shared__ float Bs[2][BK * LDB];

    const int tid  = threadIdx.x;
    const int lane = tid & 31;
    const int wave = tid >> 5;      // 0..7
    const int wm   = wave >> 1;     // 0..3 -> 32-row strip
    const int wn   = wave & 1;      // 0..1 -> 64-col strip
    const int l15  = lane & 15;
    const int lhi  = lane >> 4;     // 0 or 1 (half-wave)

    const int bm = blockIdx.y * BM;
    const int bn = blockIdx.x * BN;

    // Global->LDS copy coordinates (each thread: 2 float4 of A, 2 float4 of B)
    const int ar = tid >> 2;           // 0..63   (A row within tile, +64 for 2nd)
    const int ac = (tid & 3) * 4;      // 0,4,8,12 (A col within tile)
    const int br = tid >> 5;           // 0..7    (B row within tile, +8 for 2nd)
    const int bc = (tid & 31) * 4;     // 0..124  (B col within tile)

    const float* xg = x   + (size_t)(bm + ar) * K_TOT + ac;
    const float* rg = rot + (size_t)br * N_TOT + bn + bc;

    v8f acc[2][4];
#pragma unroll
    for (int i = 0; i < 2; ++i)
#pragma unroll
        for (int j = 0; j < 4; ++j)
            acc[i][j] = (v8f){};

    const int KT = K_TOT / BK;   // 256 K-tiles

    // ---- WMMA compute over one LDS-resident K-tile ----
    auto compute_tile = [&](const float* Ab, const float* Bb) {
#pragma unroll
        for (int kk = 0; kk < BK; kk += 4) {
            // A fragments: 16x4 f32; lanes 0-15: K=kk..kk+1, lanes 16-31: K=kk+2..kk+3
            v2f a[2];
#pragma unroll
            for (int mi = 0; mi < 2; ++mi) {
                const int row = wm * 32 + mi * 16 + l15;
                a[mi] = *(const v2f*)&Ab[row * LDA + kk + 2 * lhi];
            }
            const int krow = kk + 2 * lhi;
#pragma unroll
            for (int ni = 0; ni < 4; ++ni) {
                // B fragment: 4x16 f32; VGPR0 = rows {kk, kk+2}, VGPR1 = rows {kk+1, kk+3}
                const int col = wn * 64 + ni * 16 + l15;
                v2f b;
                b.x = Bb[krow * LDB + col];
                b.y = Bb[(krow + 1) * LDB + col];
#pragma unroll
                for (int mi = 0; mi < 2; ++mi) {
                    acc[mi][ni] = __builtin_amdgcn_wmma_f32_16x16x4_f32(
                        false, a[mi], false, b, (short)0, acc[mi][ni], false, false);
                }
            }
        }
    };

#if USE_ASYNC_LDS
    // ---- CDNA5 async copy path: global -> LDS without VGPR staging ----
    auto issue_stage = [&](int sbuf, int kt) {
        const float* xs = xg + (size_t)kt * BK;
        const float* rs = rg + (size_t)kt * BK * N_TOT;
        __builtin_amdgcn_global_load_async_to_lds_b128(
            (as1_v4i*)(as1_void*)xs,
            (as3_v4i*)(as3_void*)&As[sbuf][ar * LDA + ac], 0, 0);
        __builtin_amdgcn_global_load_async_to_lds_b128(
            (as1_v4i*)(as1_void*)(xs + (size_t)64 * K_TOT),
            (as3_v4i*)(as3_void*)&As[sbuf][(ar + 64) * LDA + ac], 0, 0);
        __builtin_amdgcn_global_load_async_to_lds_b128(
            (as1_v4i*)(as1_void*)rs,
            (as3_v4i*)(as3_void*)&Bs[sbuf][br * LDB + bc], 0, 0);
        __builtin_amdgcn_global_load_async_to_lds_b128(
            (as1_v4i*)(as1_void*)(rs + (size_t)8 * N_TOT),
            (as3_v4i*)(as3_void*)&Bs[sbuf][(br + 8) * LDB + bc], 0, 0);
    };

    issue_stage(0, 0);
    __builtin_amdgcn_s_wait_asynccnt(0);
    __syncthreads();

    int buf = 0;
    for (int kt = 0; kt < KT; ++kt) {
        const bool more = (kt + 1) < KT;
        if (more) issue_stage(buf ^ 1, kt + 1);   // overlaps with compute below
        compute_tile(&As[buf][0], &Bs[buf][0]);
        if (more) {
            __builtin_amdgcn_s_wait_asynccnt(0);
            __syncthreads();
            buf ^= 1;
        }
    }
#else
    // ---- fallback: register-staged double buffering ----
    {
        float4 a0 = *(const float4*)(xg);
        float4 a1 = *(const float4*)(xg + (size_t)64 * K_TOT);
        float4 b0 = *(const float4*)(rg);
        float4 b1 = *(const float4*)(rg + (size_t)8 * N_TOT);
        *(float4*)&As[0][ar * LDA + ac]        = a0;
        *(float4*)&As[0][(ar + 64) * LDA + ac] = a1;
        *(float4*)&Bs[0][br * LDB + bc]        = b0;
        *(float4*)&Bs[0][(br + 8) * LDB + bc]  = b1;
        __syncthreads();
    }
    int buf = 0;
    for (int kt = 0; kt < KT; ++kt) {
        float4 na0, na1, nb0, nb1;
        const bool more = (kt + 1) < KT;
        if (more) {
            const float* xg2 = xg + (size_t)(kt + 1) * BK;
            const float* rg2 = rg + (size_t)(kt + 1) * BK * N_TOT;
            na0 = *(const float4*)(xg2);
            na1 = *(const float4*)(xg2 + (size_t)64 * K_TOT);
            nb0 = *(const float4*)(rg2);
            nb1 = *(const float4*)(rg2 + (size_t)8 * N_TOT);
        }
        compute_tile(&As[buf][0], &Bs[buf][0]);
        if (more) {
            const int nbuf = buf ^ 1;
            *(float4*)&As[nbuf][ar * LDA + ac]        = na0;
            *(float4*)&As[nbuf][(ar + 64) * LDA + ac] = na1;
            *(float4*)&Bs[nbuf][br * LDB + bc]        = nb0;
            *(float4*)&Bs[nbuf][(br + 8) * LDB + bc]  = nb1;
            __syncthreads();
            buf = nbuf;
        }
    }
#endif

    // ---- epilogue: scale by smooth[col], stream out (non-temporal: keep rotation in L2) ----
#pragma unroll
    for (int ni = 0; ni < 4; ++ni) {
        const int col = bn + wn * 64 + ni * 16 + l15;
        const float s = smooth[col];
#pragma unroll
        for (int mi = 0; mi < 2; ++mi) {
            const int row0 = bm + wm * 32 + mi * 16 + 8 * lhi;
#pragma unroll
            for (int e = 0; e < 8; ++e) {
                const float v = acc[mi][ni][e] * s;
                __builtin_nontemporal_store(v, &out[(size_t)(row0 + e) * N_TOT + col]);
            }
        }
    }
}

extern "C" void kernel_launch(void* const* d_in, const int* in_sizes, int n_in,
                              void* d_out, int out_size, void* d_ws, size_t ws_size,
                              hipStream_t stream) {
    const float* x      = (const float*)d_in[0];   // (4,4096,4096) f32
    const float* rot    = (const float*)d_in[1];   // (4096,4096)  f32
    const float* smooth = (const float*)d_in[2];   // (4096,)      f32
    float* out          = (float*)d_out;           // (4,4096,4096) f32

    dim3 grid(N_TOT / BN, M_TOT / BM);             // (32, 128)
    dim3 block(256);
    rotate_smooth_gemm_f32<<<grid, block, 0, stream>>>(x, rot, smooth, out);
}